// CausalGraphNeuralNetwork_36103495090742
// MI455X (gfx1250) — compile-verified
//
#include <hip/hip_runtime.h>

// Sizes fixed by the reference.
#define B_ 16
#define N_ 256
#define H_ 128
#define L_ 6

typedef __attribute__((ext_vector_type(2))) float v2f;
typedef __attribute__((ext_vector_type(8))) float v8f;

// fp32 WMMA: D(16x16,f32) = A(16x4,f32) * B(4x16,f32) + C
// 8-arg form: (neg_a, A, neg_b, B, c_mod, C, reuse_a, reuse_b)
__device__ __forceinline__ v8f wmma_k4(v2f a, v2f b, v8f c) {
  return __builtin_amdgcn_wmma_f32_16x16x4_f32(false, a, false, b, (short)0, c,
                                               false, false);
}

// -------------------------------------------------------------------------
// h0[bn, d] = embed_table[ids[bn], d]
__global__ __launch_bounds__(256) void embed_k(const int* __restrict__ ids,
                                               const float* __restrict__ table,
                                               float* __restrict__ h) {
  const int idx = blockIdx.x * 256 + threadIdx.x;  // over B*N*H
  const int bn = idx >> 7;                          // /H
  const int d = idx & (H_ - 1);
  h[idx] = table[ids[bn] * H_ + d];
}

// -------------------------------------------------------------------------
// One GNN layer: hout = relu((A_b @ hin_b) @ W + bias)
// Grid: (N/16, B). Block: 256 threads = 8 waves. Wave w owns cols [16w,16w+16).
#define LA_S 260  // 16 x 256 A-tile, pad: 260 % 64 == 4 -> conflict-free A-frag
#define LH_S 132  // 256 x 128 h_b, pad
#define LG_S 132  // 16 x 128 agg tile, pad: 132 % 64 == 4 -> conflict-free

__global__ __launch_bounds__(256) void gnn_layer_k(
    const float* __restrict__ adj, const float* __restrict__ hin,
    const float* __restrict__ W, const float* __restrict__ bias,
    float* __restrict__ hout) {
  __shared__ float lA[16 * LA_S];
  __shared__ float lH[N_ * LH_S];
  __shared__ float lAgg[16 * LG_S];

  const int t = threadIdx.x;
  const int b = blockIdx.y;
  const int iBase = blockIdx.x * 16;

  const float* Ab = adj + (size_t)b * N_ * N_ + (size_t)iBase * N_;
  const float* Hb = hin + (size_t)b * N_ * H_;

  // Stage A row-tile (16 x 256), coalesced.
#pragma unroll
  for (int r = 0; r < 16; ++r) lA[r * LA_S + t] = Ab[r * N_ + t];
  // Stage full h_b (256 x 128), coalesced.
#pragma unroll 4
  for (int s = 0; s < (N_ * H_) / 256; ++s) {
    const int idx = s * 256 + t;
    lH[(idx >> 7) * LH_S + (idx & 127)] = Hb[idx];
  }
  __syncthreads();

  const int lane = t & 31;
  const int half = lane >> 4;  // lane-half per ISA WMMA layout
  const int lr = lane & 15;
  const int nB = (t >> 5) * 16;  // wave id * 16 = column tile base

  __builtin_prefetch(W, 0, 0);  // global_prefetch_b8 into L2 for phase 2

  // Phase 1: agg_tile[16,16] = A_tile[16,256] @ h_b[256, nB:nB+16]
  v8f acc = {0.f, 0.f, 0.f, 0.f, 0.f, 0.f, 0.f, 0.f};
#pragma unroll 8
  for (int k0 = 0; k0 < N_; k0 += 4) {
    const int ka = k0 + half * 2;
    v2f a = {lA[lr * LA_S + ka], lA[lr * LA_S + ka + 1]};
    v2f bb = {lH[ka * LH_S + nB + lr], lH[(ka + 1) * LH_S + nB + lr]};
    acc = wmma_k4(a, bb, acc);
  }
#pragma unroll
  for (int r = 0; r < 8; ++r) lAgg[(r + half * 8) * LG_S + nB + lr] = acc[r];
  __syncthreads();

  // Phase 2: out[16,16] = relu(agg[16,128] @ W[:, nB:nB+16] + bias)
  v8f o = {0.f, 0.f, 0.f, 0.f, 0.f, 0.f, 0.f, 0.f};
#pragma unroll 8
  for (int k0 = 0; k0 < H_; k0 += 4) {
    const int ka = k0 + half * 2;
    v2f a = {lAgg[lr * LG_S + ka], lAgg[lr * LG_S + ka + 1]};
    v2f bb = {W[ka * H_ + nB + lr], W[(ka + 1) * H_ + nB + lr]};
    o = wmma_k4(a, bb, o);
  }
  const float bv = bias[nB + lr];
  float* Ob = hout + (size_t)b * N_ * H_ + (size_t)iBase * H_;
#pragma unroll
  for (int r = 0; r < 8; ++r) {
    const float v = o[r] + bv;
    Ob[(r + half * 8) * H_ + nB + lr] = v > 0.f ? v : 0.f;
  }
}

// -------------------------------------------------------------------------
// Generic out[M,128] = act(in[M,128] @ W[128,128] + bias), M = B*N
// Grid: M/16 blocks. Block: 256 threads = 8 waves.
#define LI_S 132

__global__ __launch_bounds__(256) void rowgemm_k(
    const float* __restrict__ in, const float* __restrict__ W,
    const float* __restrict__ bias, int doRelu, float* __restrict__ out) {
  __shared__ float lIn[16 * LI_S];
  const int t = threadIdx.x;
  const int mBase = blockIdx.x * 16;
  const float* Ib = in + (size_t)mBase * H_;
#pragma unroll
  for (int s = 0; s < 8; ++s) {
    const int idx = s * 256 + t;
    lIn[(idx >> 7) * LI_S + (idx & 127)] = Ib[idx];
  }
  __syncthreads();

  const int lane = t & 31;
  const int half = lane >> 4;
  const int lr = lane & 15;
  const int nB = (t >> 5) * 16;

  v8f o = {0.f, 0.f, 0.f, 0.f, 0.f, 0.f, 0.f, 0.f};
#pragma unroll 8
  for (int k0 = 0; k0 < H_; k0 += 4) {
    const int ka = k0 + half * 2;
    v2f a = {lIn[lr * LI_S + ka], lIn[lr * LI_S + ka + 1]};
    v2f bb = {W[ka * H_ + nB + lr], W[(ka + 1) * H_ + nB + lr]};
    o = wmma_k4(a, bb, o);
  }
  const float bv = bias ? bias[nB + lr] : 0.f;
  float* Ob = out + (size_t)mBase * H_;
#pragma unroll
  for (int r = 0; r < 8; ++r) {
    float v = o[r] + bv;
    if (doRelu) v = v > 0.f ? v : 0.f;
    Ob[(r + half * 8) * H_ + nB + lr] = v;
  }
}

// -------------------------------------------------------------------------
// Fused edge scorer: never materializes the (B,N,N,H) tensor.
// edge[b,i,j] = sigmoid( sum_h relu(Ai[b,i,h]+b1[h]+Bj[b,j,h]) * W2[h] + b2 )
//              masked on the diagonal.
// Grid: (N/16, B). Block: 256 threads; thread t handles column j = t.
#define LBJ_S 129  // stride-129 pad: lane j -> distinct LDS banks

__global__ __launch_bounds__(256) void edge_k(
    const float* __restrict__ Ai, const float* __restrict__ Bj,
    const float* __restrict__ b1, const float* __restrict__ W2,
    const float* __restrict__ b2p, float* __restrict__ eout) {
  __shared__ float lBj[N_ * LBJ_S];
  __shared__ float lAi[16 * H_];
  __shared__ float lW2[H_];

  const int t = threadIdx.x;
  const int b = blockIdx.y;
  const int iBase = blockIdx.x * 16;

  const float* Bb = Bj + (size_t)b * N_ * H_;
#pragma unroll 4
  for (int s = 0; s < (N_ * H_) / 256; ++s) {
    const int idx = s * 256 + t;
    lBj[(idx >> 7) * LBJ_S + (idx & 127)] = Bb[idx];
  }
  const float* Aib = Ai + (size_t)b * N_ * H_ + (size_t)iBase * H_;
#pragma unroll
  for (int s = 0; s < 8; ++s) {
    const int idx = s * 256 + t;
    lAi[idx] = Aib[idx] + b1[idx & 127];  // fold edge_b1 once
  }
  if (t < H_) lW2[t] = W2[t];
  __syncthreads();

  const float b2 = b2p[0];
  const int j = t;
  float* Eb = eout + (size_t)b * N_ * N_ + (size_t)iBase * N_;
  const float* bj = &lBj[j * LBJ_S];
  for (int ii = 0; ii < 16; ++ii) {
    const float* ai = &lAi[ii * H_];  // lane-uniform -> LDS broadcast
    float acc = 0.f;
#pragma unroll 8
    for (int h = 0; h < H_; ++h) {
      float v = ai[h] + bj[h];
      v = v > 0.f ? v : 0.f;
      acc = fmaf(v, lW2[h], acc);
    }
    const float logit = acc + b2;
    float p = 1.f / (1.f + __expf(-logit));
    if (iBase + ii == j) p = 0.f;  // (1 - eye) mask
    Eb[ii * N_ + j] = p;
  }
}

// -------------------------------------------------------------------------
extern "C" void kernel_launch(void* const* d_in, const int* in_sizes, int n_in,
                              void* d_out, int out_size, void* d_ws,
                              size_t ws_size, hipStream_t stream) {
  (void)in_sizes; (void)n_in; (void)out_size; (void)ws_size;
  const int* ids = (const int*)d_in[0];
  const float* adj = (const float*)d_in[1];
  const float* table = (const float*)d_in[2];
  const float* convW = (const float*)d_in[3];
  const float* convb = (const float*)d_in[4];
  const float* eW1 = (const float*)d_in[5];
  const float* eb1 = (const float*)d_in[6];
  const float* eW2 = (const float*)d_in[7];
  const float* eb2 = (const float*)d_in[8];
  const float* mW1 = (const float*)d_in[9];
  const float* mb1 = (const float*)d_in[10];
  const float* mW2 = (const float*)d_in[11];
  const float* mb2 = (const float*)d_in[12];

  const size_t HN = (size_t)B_ * N_ * H_;  // 524288
  float* hA = (float*)d_ws;
  float* hB = hA + HN;
  float* AiBuf = hB + HN;
  float* BjBuf = AiBuf + HN;
  float* mechT = BjBuf + HN;

  float* out_h = (float*)d_out;
  float* out_e = out_h + HN;                       // (B,N,N)
  float* out_m = out_e + (size_t)B_ * N_ * N_;     // (B,N,H)

  embed_k<<<(int)(HN / 256), 256, 0, stream>>>(ids, table, hA);

  const float* cur = hA;
  for (int l = 0; l < L_; ++l) {
    float* dst = (l == L_ - 1) ? out_h : ((l & 1) ? hA : hB);
    gnn_layer_k<<<dim3(N_ / 16, B_), 256, 0, stream>>>(
        adj, cur, convW + (size_t)l * H_ * H_, convb + (size_t)l * H_, dst);
    cur = dst;
  }

  const int MB = (B_ * N_) / 16;  // 256 row-tiles
  rowgemm_k<<<MB, 256, 0, stream>>>(out_h, eW1, nullptr, 0, AiBuf);
  rowgemm_k<<<MB, 256, 0, stream>>>(out_h, eW1 + H_ * H_, nullptr, 0, BjBuf);
  rowgemm_k<<<MB, 256, 0, stream>>>(out_h, mW1, mb1, 1, mechT);
  rowgemm_k<<<MB, 256, 0, stream>>>(mechT, mW2, mb2, 0, out_m);

  edge_k<<<dim3(N_ / 16, B_), 256, 0, stream>>>(AiBuf, BjBuf, eb1, eW2, eb2,
                                                out_e);
}